// TensorProduct_83648783057673
// MI455X (gfx1250) — compile-verified
//
#include <hip/hip_runtime.h>

// ---------------------------------------------------------------------------
// CG tensor product, irreps (1,3,5), MUL=32, Z samples.
// Factored form: out[z] (32x9) = x1[z] (32x9) @ W[z] (9x9),
//   W[z][f,g] = sum_j T[j,f,g] * x2[z][j],  T = fixed scatter of cg.
// Main matmul runs on v_wmma_f32_16x16x4_f32 (K padded 9->12, N padded 9->16).
// ---------------------------------------------------------------------------

typedef float v2f __attribute__((ext_vector_type(2)));
typedef float v8f __attribute__((ext_vector_type(8)));

#define NPATH 15
#define TF 12            // f (K) dim padded 9->12
#define TG 16            // g (N) dim padded 9->16
#define TSIZE (9 * TF * TG)   // 1728 floats

__device__ __constant__ int c_path[NPATH][3] = {
    {0,0,0},{0,1,1},{0,2,2},
    {1,0,1},{1,1,0},{1,1,1},{1,1,2},{1,2,1},{1,2,2},
    {2,0,2},{2,1,1},{2,1,2},{2,2,0},{2,2,1},{2,2,2}};
__device__ __constant__ int c_foff[3] = {0, 1, 4};   // cumsum of dims {1,3,5}

// feature f (0..8) -> flat offset within a z-row of x1/out (u-major blocks)
__device__ __forceinline__ int feat_off(int u, int f) {
    return (f == 0) ? u
         : (f < 4)  ? (32  + u * 3 + (f - 1))
                    : (128 + u * 5 + (f - 4));
}

__global__ void zeroT_kernel(float* __restrict__ T) {
    int i = blockIdx.x * blockDim.x + threadIdx.x;
    if (i < TSIZE) T[i] = 0.0f;
}

// scatter cg (15 x 5 x 5 x 5) into dense padded T[j][f][g]
__global__ void buildT_kernel(const float* __restrict__ cg, float* __restrict__ T) {
    int t = blockIdx.x * blockDim.x + threadIdx.x;
    if (t >= NPATH * 125) return;
    int p = t / 125, r = t % 125;
    int i = r / 25, j = (r % 25) / 5, k = r % 5;
    int i1 = c_path[p][0], i2 = c_path[p][1], i3 = c_path[p][2];
    if (i < 2 * i1 + 1 && j < 2 * i2 + 1 && k < 2 * i3 + 1) {
        int f  = c_foff[i1] + i;
        int g  = c_foff[i3] + k;
        int jj = c_foff[i2] + j;
        T[(jj * TF + f) * TG + g] = cg[t];
    }
}

__global__ __launch_bounds__(256) void tp_wmma_kernel(
    const float* __restrict__ x1, const float* __restrict__ x2,
    const float* __restrict__ T, float* __restrict__ out, int Z) {
    const int lane = threadIdx.x & 31;
    const int h    = lane >> 4;       // lane half
    const int n    = lane & 15;       // N column (B/D), M row base (A)
    const int wid  = (int)((blockIdx.x * blockDim.x + threadIdx.x) >> 5);
    const int nw   = (int)((gridDim.x * blockDim.x) >> 5);

    // q = 0..5 : chunk c = q>>1, operand VGPR v = q&1 ; K/f = 4c + 2h + v
    // (same selector for this lane's A element and B element)
    int fsel[6];
    float Treg[6][9];                 // per-lane slice of T, kept in VGPRs
    #pragma unroll
    for (int q = 0; q < 6; q++) {
        int c = q >> 1, v = q & 1;
        int f = 4 * c + 2 * h + v;    // 0..11 (>=9 are zero padding in T)
        fsel[q] = f;
        #pragma unroll
        for (int j = 0; j < 9; j++)
            Treg[q][j] = T[(j * TF + f) * TG + n];
    }

    for (int z = wid; z < Z; z += nw) {
        const float* x1z = x1 + (size_t)z * 288;
        const float* x2z = x2 + (size_t)z * 9;
        float*       oz  = out + (size_t)z * 288;

        if (z + nw < Z)  // warm L2/L1 for next iteration's x1 row
            __builtin_prefetch(x1 + (size_t)(z + nw) * 288 + lane * 9, 0, 0);

        float xv[9];
        #pragma unroll
        for (int j = 0; j < 9; j++) xv[j] = x2z[j];

        // B operand values: W[f, n] for this lane's six (c,v) slots
        float bq[6];
        #pragma unroll
        for (int q = 0; q < 6; q++) {
            float s = 0.0f;
            #pragma unroll
            for (int j = 0; j < 9; j++) s = fmaf(Treg[q][j], xv[j], s);
            bq[q] = s;
        }

        #pragma unroll
        for (int m = 0; m < 2; m++) {           // M tiles: u 0..15, 16..31
            const int u = n + 16 * m;           // this lane's A row
            float a[6];
            #pragma unroll
            for (int q = 0; q < 6; q++) {       // A operand values
                int f = fsel[q];
                float val = 0.0f;
                if (f < 9) val = x1z[feat_off(u, f)];
                a[q] = val;
            }

            v8f acc = {};
            #pragma unroll
            for (int c = 0; c < 3; c++) {       // K chunks of 4
                v2f A = {a[2 * c], a[2 * c + 1]};
                v2f B = {bq[2 * c], bq[2 * c + 1]};
                acc = __builtin_amdgcn_wmma_f32_16x16x4_f32(
                    false, A, false, B, (short)0, acc, false, false);
            }

            if (n < 9) {                        // store 9 real columns
                #pragma unroll
                for (int r = 0; r < 8; r++) {
                    int uo = r + 8 * h + 16 * m;
                    oz[feat_off(uo, n)] = acc[r];
                }
            }
        }
    }
}

extern "C" void kernel_launch(void* const* d_in, const int* in_sizes, int n_in,
                              void* d_out, int out_size, void* d_ws, size_t ws_size,
                              hipStream_t stream) {
    const float* x1 = (const float*)d_in[0];   // Z x 288
    const float* x2 = (const float*)d_in[1];   // Z x 9
    const float* cg = (const float*)d_in[2];   // 15 x 5 x 5 x 5
    float* out = (float*)d_out;
    float* T   = (float*)d_ws;                 // 1728 floats = 6912 B
    const int Z = in_sizes[0] / 288;

    zeroT_kernel<<<(TSIZE + 255) / 256, 256, 0, stream>>>(T);
    buildT_kernel<<<(NPATH * 125 + 255) / 256, 256, 0, stream>>>(cg, T);

    int blocks = 1024;                          // 8192 waves, ~49 z each
    tp_wmma_kernel<<<blocks, 256, 0, stream>>>(x1, x2, T, out, Z);
}